// SUEPNet_90838558310842
// MI455X (gfx1250) — compile-verified
//
#include <hip/hip_runtime.h>
#include <hip/hip_bf16.h>
#include <math.h>

typedef __attribute__((ext_vector_type(2))) float v2f;
typedef __attribute__((ext_vector_type(8))) float v8f;

#define B_EV 256
#define NPERT 512
#define KNN 8
#define HD 16
#define HS 18                    // padded h_lds row stride (words): even (b64-aligned), 32 banks
#define DS 33                    // padded dstripe row stride (words): coprime with 64 banks
#define CHUNK 32
#define NCT (CHUNK / 16)         // 2 col tiles per chunk
#define NCHUNK (NPERT / CHUNK)   // 16 chunks

__device__ __forceinline__ float elu1(float x) {
    return x > 0.0f ? x : (__expf(x) - 1.0f);
}

// ---------------- pointwise encoder: x[4] -> elu(W1) -> elu(W2) -> h[16] ----------------
__global__ __launch_bounds__(256) void suep_encode_kernel(
    const float* __restrict__ x_pf,
    const float* __restrict__ W1, const float* __restrict__ b1,
    const float* __restrict__ W2, const float* __restrict__ b2,
    float* __restrict__ h_out)
{
    int n = blockIdx.x * blockDim.x + threadIdx.x;
    float x[4];
    #pragma unroll
    for (int i = 0; i < 4; ++i) x[i] = x_pf[n * 4 + i];
    float h1[HD];
    #pragma unroll
    for (int o = 0; o < HD; ++o) {
        float s = b1[o];
        #pragma unroll
        for (int i = 0; i < 4; ++i) s += x[i] * W1[i * HD + o];
        h1[o] = elu1(s);
    }
    #pragma unroll
    for (int o = 0; o < HD; ++o) {
        float s = b2[o];
        #pragma unroll
        for (int j = 0; j < HD; ++j) s += h1[j] * W2[j * HD + o];
        h_out[(size_t)n * HD + o] = elu1(s);
    }
}

// ---------------- DynamicEdgeConv: one block per event, 8 waves ----------------
__global__ __launch_bounds__(256) void suep_edgeconv_kernel(
    const float* __restrict__ h_in, float* __restrict__ h_out,
    const float* __restrict__ Wc, const float* __restrict__ bc)
{
    __shared__ float h_lds[NPERT * HS];        // 36 KB: event features (padded rows)
    __shared__ float sq_lds[NPERT];            // 2 KB : squared norms
    __shared__ float dstripe[NPERT * DS];      // 66 KB: d2, all rows x 32-col chunk (padded)
    __shared__ float wc_lds[2 * HD * HD];      // 2 KB : edge MLP weights
    __shared__ float bc_lds[HD];

    const int tid  = threadIdx.x;
    const int wave = tid >> 5;
    const int lane = tid & 31;
    const int hsel = lane >> 4;      // which 16-lane half
    const int lm   = lane & 15;      // lane mod 16

    const float* hg = h_in + (size_t)blockIdx.x * NPERT * HD;

    for (int i = tid; i < NPERT * HD; i += 256)
        h_lds[(i >> 4) * HS + (i & 15)] = hg[i];
    for (int i = tid; i < 2 * HD * HD; i += 256) wc_lds[i] = Wc[i];
    if (tid < HD) bc_lds[tid] = bc[tid];
    __syncthreads();
    for (int r = tid; r < NPERT; r += 256) {
        float s = 0.f;
        #pragma unroll
        for (int d = 0; d < HD; ++d) { float v = h_lds[r * HS + d]; s += v * v; }
        sq_lds[r] = s;
    }
    __syncthreads();

    // each thread owns rows tid and tid+256; running top-8 (dist, idx), sorted ascending
    float d8[2][KNN];
    int   i8[2][KNN];
    #pragma unroll
    for (int r = 0; r < 2; ++r)
        #pragma unroll
        for (int k = 0; k < KNN; ++k) { d8[r][k] = 3.4e38f; i8[r][k] = 0; }

    for (int c = 0; c < NCHUNK; ++c) {
        // ---- load this wave's B-tiles once per chunk (shared by all 4 row-tiles) ----
        v2f Bt[NCT][4];
        #pragma unroll
        for (int ct = 0; ct < NCT; ++ct) {
            const int colrow = (c * CHUNK + ct * 16 + lm) * HS + 2 * hsel;
            #pragma unroll
            for (int kc = 0; kc < 4; ++kc)
                Bt[ct][kc] = *(const v2f*)&h_lds[colrow + 4 * kc];
        }
        // ---- 4 row-tiles per wave, uniform trip counts (EXEC stays all-ones) ----
        #pragma unroll
        for (int i = 0; i < 4; ++i) {
            const int rowbase = (wave * 4 + i) * 16;
            v2f At[4];
            {
                const int arow = (rowbase + lm) * HS + 2 * hsel;
                #pragma unroll
                for (int kc = 0; kc < 4; ++kc)
                    At[kc] = *(const v2f*)&h_lds[arow + 4 * kc];
            }
            float sqr[8];
            #pragma unroll
            for (int r = 0; r < 8; ++r) sqr[r] = sq_lds[rowbase + r + 8 * hsel];
            #pragma unroll
            for (int ct = 0; ct < NCT; ++ct) {
                v8f acc = {0.f, 0.f, 0.f, 0.f, 0.f, 0.f, 0.f, 0.f};
                #pragma unroll
                for (int kc = 0; kc < 4; ++kc)
                    acc = __builtin_amdgcn_wmma_f32_16x16x4_f32(
                            false, At[kc], false, Bt[ct][kc], (short)0, acc, false, false);
                const float sqc = sq_lds[c * CHUNK + ct * 16 + lm];
                #pragma unroll
                for (int r = 0; r < 8; ++r) {
                    const int row = rowbase + r + 8 * hsel;
                    dstripe[row * DS + ct * 16 + lm] = sqr[r] + sqc - 2.0f * acc[r];
                }
            }
        }
        __syncthreads();

        // ---- top-8 insertion over this chunk (self included, as in reference) ----
        #pragma unroll
        for (int r = 0; r < 2; ++r) {
            const int row = tid + r * 256;
            const float* dr = &dstripe[row * DS];
            for (int j = 0; j < CHUNK; ++j) {
                const float d = dr[j];
                if (d < d8[r][KNN - 1]) {           // rare after warm-up
                    d8[r][KNN - 1] = d; i8[r][KNN - 1] = c * CHUNK + j;
                    #pragma unroll
                    for (int q = KNN - 1; q > 0; --q) {   // branch-free compare-swap chain
                        const bool sw = d8[r][q] < d8[r][q - 1];
                        const float td = sw ? d8[r][q] : d8[r][q - 1];
                        d8[r][q]     = sw ? d8[r][q - 1] : d8[r][q];
                        const int ti = sw ? i8[r][q] : i8[r][q - 1];
                        i8[r][q]     = sw ? i8[r][q - 1] : i8[r][q];
                        d8[r][q - 1] = td; i8[r][q - 1] = ti;
                    }
                }
            }
        }
        __syncthreads();
    }

    // ---- edge MLP + max aggregate ----
    #pragma unroll
    for (int r = 0; r < 2; ++r) {
        const int row = tid + r * 256;
        float hi[HD];
        #pragma unroll
        for (int d = 0; d < HD; ++d) hi[d] = h_lds[row * HS + d];
        // hoist the x_i part: base[o] = bc[o] + hi . Wc[0:16, o]
        float base[HD];
        #pragma unroll
        for (int o = 0; o < HD; ++o) {
            float s = bc_lds[o];
            #pragma unroll
            for (int d = 0; d < HD; ++d) s += hi[d] * wc_lds[d * HD + o];
            base[o] = s;
        }
        float agg[HD];
        #pragma unroll
        for (int o = 0; o < HD; ++o) agg[o] = -3.4e38f;
        #pragma unroll
        for (int k = 0; k < KNN; ++k) {
            const int j = i8[r][k];
            float dlt[HD];
            #pragma unroll
            for (int d = 0; d < HD; ++d) dlt[d] = h_lds[j * HS + d] - hi[d];
            #pragma unroll
            for (int o = 0; o < HD; ++o) {
                float s = base[o];
                #pragma unroll
                for (int d = 0; d < HD; ++d) s += dlt[d] * wc_lds[(HD + d) * HD + o];
                agg[o] = fmaxf(agg[o], elu1(s));
            }
        }
        float* og = h_out + ((size_t)blockIdx.x * NPERT + row) * HD;
        #pragma unroll
        for (int o = 0; o < HD; ++o) og[o] = agg[o];
    }
}

// ---------------- per-event mean pool + head MLP (16->8->4->1) ----------------
__global__ __launch_bounds__(256) void suep_head_kernel(
    const float* __restrict__ f2,
    const float* __restrict__ Wo1, const float* __restrict__ bo1,
    const float* __restrict__ Wo2, const float* __restrict__ bo2,
    const float* __restrict__ Wo3, const float* __restrict__ bo3,
    float* __restrict__ out, int out_size)
{
    __shared__ float part[256 * HD];   // 16 KB
    __shared__ float pooled[HD];
    const int tid = threadIdx.x, ev = blockIdx.x;
    const float* fg = f2 + (size_t)ev * NPERT * HD;
    #pragma unroll
    for (int o = 0; o < HD; ++o)
        part[tid * HD + o] = fg[tid * HD + o] + fg[(tid + 256) * HD + o];
    __syncthreads();
    if (tid < HD) {
        float acc = 0.f;
        for (int t = 0; t < 256; ++t) acc += part[t * HD + tid];
        pooled[tid] = acc * (1.0f / (float)NPERT);
    }
    __syncthreads();
    if (tid == 0) {
        float o1[8];
        #pragma unroll
        for (int o = 0; o < 8; ++o) {
            float v = bo1[o];
            #pragma unroll
            for (int d = 0; d < HD; ++d) v += pooled[d] * Wo1[d * 8 + o];
            o1[o] = elu1(v);
        }
        float o2[4];
        #pragma unroll
        for (int o = 0; o < 4; ++o) {
            float v = bo2[o];
            #pragma unroll
            for (int d = 0; d < 8; ++d) v += o1[d] * Wo2[d * 4 + o];
            o2[o] = elu1(v);
        }
        float v = bo3[0];
        #pragma unroll
        for (int d = 0; d < 4; ++d) v += o2[d] * Wo3[d];
        out[ev] = v;
        if (out_size >= 2 * B_EV) out[B_EV + ev] = (float)ev;  // second tuple output: arange(B)
    }
}

extern "C" void kernel_launch(void* const* d_in, const int* in_sizes, int n_in,
                              void* d_out, int out_size, void* d_ws, size_t ws_size,
                              hipStream_t stream)
{
    const float* x_pf = (const float*)d_in[0];
    // d_in[1] = batch_pf (contiguous equal-size events; unused)
    const float* W1  = (const float*)d_in[2];
    const float* b1  = (const float*)d_in[3];
    const float* W2  = (const float*)d_in[4];
    const float* b2  = (const float*)d_in[5];
    const float* Wc  = (const float*)d_in[6];
    const float* bc  = (const float*)d_in[7];
    const float* Wo1 = (const float*)d_in[8];
    const float* bo1 = (const float*)d_in[9];
    const float* Wo2 = (const float*)d_in[10];
    const float* bo2 = (const float*)d_in[11];
    const float* Wo3 = (const float*)d_in[12];
    const float* bo3 = (const float*)d_in[13];

    const int N = B_EV * NPERT;
    float* h0 = (float*)d_ws;                  // encoder output   [N,16]
    float* h1 = h0 + (size_t)N * HD;           // edgeconv 1 out   [N,16]
    float* h2 = h1 + (size_t)N * HD;           // edgeconv 2 out   [N,16]

    suep_encode_kernel<<<N / 256, 256, 0, stream>>>(x_pf, W1, b1, W2, b2, h0);
    suep_edgeconv_kernel<<<B_EV, 256, 0, stream>>>(h0, h1, Wc, bc);
    suep_edgeconv_kernel<<<B_EV, 256, 0, stream>>>(h1, h2, Wc, bc);
    suep_head_kernel<<<B_EV, 256, 0, stream>>>(h2, Wo1, bo1, Wo2, bo2, Wo3, bo3,
                                               (float*)d_out, out_size);
}